// TrainableMovingAverage1d_22557168239332
// MI455X (gfx1250) — compile-verified
//
#include <hip/hip_runtime.h>

typedef __attribute__((ext_vector_type(2))) float v2f;
typedef __attribute__((ext_vector_type(8))) float v8f;

#define ROWS      4096
#define COLS      8192
#define KTAPS     512
#define OUTC      (COLS - KTAPS)     // 7680 output columns
#define KK_TOT    528                // covers kk = 0..527 (= 511 + 15 + 1, rounded to x4)
#define LDS_STRIDE 532               // pad: 532 % 64 = 20 -> conflict-free lane spread
#define WAVES_PER_BLOCK 8
#define ROWTILES_PER_WAVE 2
#define BLOCK_ROWS (WAVES_PER_BLOCK * ROWTILES_PER_WAVE * 16)   // 256 rows per workgroup

__global__ __launch_bounds__(256)
void tma1d_wmma_f32(const float* __restrict__ x,
                    const float* __restrict__ w,
                    float* __restrict__ out) {
    // Toeplitz band of w, transposed: bT[n][kk] = w[kk - n], zero outside [0,512)
    __shared__ float bT[16 * LDS_STRIDE];

    const int tid = threadIdx.x;
    for (int e = tid; e < 16 * LDS_STRIDE; e += 256) {
        const int n  = e / LDS_STRIDE;
        const int kk = e - n * LDS_STRIDE;
        const int t  = kk - n;
        bT[e] = (t >= 0 && t < KTAPS && kk < KK_TOT) ? w[t] : 0.0f;
    }
    __syncthreads();

    const int wave  = tid >> 5;      // 0..7
    const int lane  = tid & 31;
    const int m     = lane & 15;     // A row-in-tile / B,D column N
    const int khalf = lane >> 4;     // 0/1: which K half-pair this lane holds

    const int i0 = blockIdx.y * BLOCK_ROWS + wave * (ROWTILES_PER_WAVE * 16);
    const int j0 = blockIdx.x * 16;                       // first output column

    // A fragment for chunk kk0, row tile r: x[i0 + r*16 + m, j0 + kk0 + khalf*2 + {0,1}]
    const float* aptr0 = x  + (size_t)(i0 + m) * COLS + j0 + khalf * 2;
    const float* aptr1 = aptr0 + (size_t)16 * COLS;
    // B fragment for chunk kk0: bT[m][kk0 + khalf*2 + {0,1}]  (shared by both row tiles)
    const float* bptr  = bT + m * LDS_STRIDE + khalf * 2;

    v8f acc0 = {};
    v8f acc1 = {};
    #pragma unroll 4
    for (int kk0 = 0; kk0 < KK_TOT; kk0 += 4) {
        v2f b  = *(const v2f*)(bptr  + kk0);
        v2f a0 = *(const v2f*)(aptr0 + kk0);
        v2f a1 = *(const v2f*)(aptr1 + kk0);
        // D = A(16x4,f32) * B(4x16,f32) + C   -> v_wmma_f32_16x16x4_f32
        acc0 = __builtin_amdgcn_wmma_f32_16x16x4_f32(
                   false, a0, false, b, (short)0, acc0, false, false);
        acc1 = __builtin_amdgcn_wmma_f32_16x16x4_f32(
                   false, a1, false, b, (short)0, acc1, false, false);
    }

    // D layout: lane L, VGPR r -> row M = r + 8*(L/16), col N = L%16
    float* obase0 = out + (size_t)(i0 + khalf * 8) * OUTC + j0 + m;
    float* obase1 = obase0 + (size_t)16 * OUTC;
    #pragma unroll
    for (int r = 0; r < 8; ++r) {
        obase0[(size_t)r * OUTC] = acc0[r];
        obase1[(size_t)r * OUTC] = acc1[r];
    }
}

extern "C" void kernel_launch(void* const* d_in, const int* in_sizes, int n_in,
                              void* d_out, int out_size, void* d_ws, size_t ws_size,
                              hipStream_t stream) {
    (void)in_sizes; (void)n_in; (void)out_size; (void)d_ws; (void)ws_size;
    const float* x = (const float*)d_in[0];   // (4096, 8192) f32
    const float* w = (const float*)d_in[1];   // (512,) f32
    float* out = (float*)d_out;               // (4096, 7680) f32

    dim3 grid(OUTC / 16, ROWS / BLOCK_ROWS);  // (480, 16)
    tma1d_wmma_f32<<<grid, 256, 0, stream>>>(x, w, out);
}